// SpanRepresentation_9543417331986
// MI455X (gfx1250) — compile-verified
//
#include <hip/hip_runtime.h>
#include <cstdint>

// Problem constants (match reference)
#define BB    8
#define SS    2048
#define HH    768
#define NN    512
#define MAXW  30
#define WPAD  32            // span width padded to two 16-token WMMA tiles
#define ROWS  772           // LDS row stride (768+4 floats): bank-conflict-free A loads, rows stay 16B aligned
#define H4    (HH / 4)      // 192 float4 per embedding row
#define THREADS 256

typedef __attribute__((ext_vector_type(2))) float v2f;
typedef __attribute__((ext_vector_type(8))) float v8f;

__global__ __launch_bounds__(THREADS)
void span_repr_kernel(const float* __restrict__ emb,
                      const float* __restrict__ w,
                      const int*   __restrict__ starts,
                      const int*   __restrict__ lengths,
                      float*       __restrict__ out)
{
    extern __shared__ float smem[];
    float* sE      = smem;                 // WPAD x ROWS span tile
    float* sWv     = smem + WPAD * ROWS;   // 768 floats of w
    float* sScores = sWv + HH;             // 32 scores
    float* sWgt    = sScores + 32;         // 32 softmax weights

    const int sid   = blockIdx.x;          // b*N + n
    const int tid   = threadIdx.x;
    const int batch = sid >> 9;            // / 512
    const float* ebase = emb + (size_t)batch * (SS * HH);
    const int start = starts[sid];
    const int len   = lengths[sid];        // inclusive end offset, 0..29

    // ---- Phase A: stage w and the padded span tile into LDS (128-bit, coalesced) ----
    for (int i = tid; i < HH; i += THREADS) sWv[i] = w[i];
    const float4* e4 = (const float4*)ebase;
    for (int idx = tid; idx < WPAD * H4; idx += THREADS) {
        int t = idx / H4;
        int c = idx - t * H4;
        int row = start + (t < MAXW ? t : (MAXW - 1));   // pad rows 30,31 (masked out later)
        *(float4*)&sE[t * ROWS + c * 4] = e4[row * H4 + c];
    }
    __syncthreads();

    const size_t obase = (size_t)sid * (3 * HH);
    const int wave = tid >> 5;
    const int lane = tid & 31;

    if (wave < 2) {
        // ---- Phase B (waves 0-1): scores via V_WMMA_F32_16X16X4_F32 ----
        // A(16x4) f32: lanes 0-15 hold K=0,1 ; lanes 16-31 hold K=2,3 of row M = lane&15.
        // B(4x16) f32: VGPR0 = row K0 (lanes 0-15) / K2 (lanes 16-31), VGPR1 = K1 / K3.
        // B columns all carry the same w chunk -> D[m][*] = score[tile*16+m].
        const int half = lane >> 4;
        const int m    = lane & 15;
        const float* Ab = sE + (wave * 16 + m) * ROWS + 2 * half;
        const float* Bb = sWv + 2 * half;
        v8f acc0 = {}; v8f acc1 = {};
        for (int k0 = 0; k0 < HH; k0 += 8) {           // two interleaved accumulation chains
            v2f a0 = { Ab[k0],     Ab[k0 + 1] };
            v2f b0 = { Bb[k0],     Bb[k0 + 1] };
            v2f a1 = { Ab[k0 + 4], Ab[k0 + 5] };
            v2f b1 = { Bb[k0 + 4], Bb[k0 + 5] };
            acc0 = __builtin_amdgcn_wmma_f32_16x16x4_f32(false, a0, false, b0,
                                                         (short)0, acc0, false, false);
            acc1 = __builtin_amdgcn_wmma_f32_16x16x4_f32(false, a1, false, b1,
                                                         (short)0, acc1, false, false);
        }
        v8f acc = acc0 + acc1;
        if (m == 0) {                                   // N=0 column: lane 0 -> M=r, lane 16 -> M=r+8
            int mb = wave * 16 + half * 8;
            #pragma unroll
            for (int r = 0; r < 8; ++r) sScores[mb + r] = acc[r];
        }
    } else {
        // ---- Phase C (waves 2-7, 192 threads): start/end segments straight out of LDS ----
        const int ct = tid - 64;                        // 0..191 -> one float4 each per segment
        float4* o4 = (float4*)(out + obase);
        o4[ct]      = *(const float4*)&sE[ct * 4];                 // start_emb (row 0)
        o4[H4 + ct] = *(const float4*)&sE[len * ROWS + ct * 4];    // end_emb   (row len)
    }
    __syncthreads();

    // ---- Phase D: masked softmax weights (bias cancels under softmax shift-invariance) ----
    if (tid < 32) {
        float mx = -3.0e38f;
        for (int t = 0; t <= len; ++t) mx = fmaxf(mx, sScores[t]);
        float den = 0.f;
        for (int t = 0; t <= len; ++t) den += __expf(sScores[t] - mx);
        sWgt[tid] = (tid <= len) ? __expf(sScores[tid] - mx) / den : 0.f;
    }
    __syncthreads();

    // ---- Phase E: attention-pooled embedding from the LDS-resident tile ----
    for (int h = tid; h < HH; h += THREADS) {
        float acc = 0.f;
        for (int t = 0; t <= len; ++t)
            acc = fmaf(sWgt[t], sE[t * ROWS + h], acc);
        out[obase + 2 * HH + h] = acc;
    }
}

extern "C" void kernel_launch(void* const* d_in, const int* in_sizes, int n_in,
                              void* d_out, int out_size, void* d_ws, size_t ws_size,
                              hipStream_t stream) {
    (void)in_sizes; (void)n_in; (void)out_size; (void)d_ws; (void)ws_size;
    const float* emb     = (const float*)d_in[0];   // (B,S,H) f32
    const float* w       = (const float*)d_in[1];   // (H,1)   f32
    // d_in[2] = bias: unused (softmax is shift-invariant; reference bias is zero anyway)
    const int*   starts  = (const int*)d_in[3];     // (B,N) i32
    const int*   lengths = (const int*)d_in[4];     // (B,N) i32
    float* out = (float*)d_out;                     // (B,N,3H) f32

    const size_t shmem = (size_t)(WPAD * ROWS + HH + 64) * sizeof(float);  // ~100 KB of 320 KB/WGP
    hipFuncSetAttribute((const void*)span_repr_kernel,
                        hipFuncAttributeMaxDynamicSharedMemorySize, (int)shmem);
    span_repr_kernel<<<BB * NN, THREADS, shmem, stream>>>(emb, w, starts, lengths, out);
}